// Lora_86182813762415
// MI455X (gfx1250) — compile-verified
//
#include <hip/hip_runtime.h>

typedef float v2f __attribute__((ext_vector_type(2)));
typedef float v4f __attribute__((ext_vector_type(4)));
typedef float v8f __attribute__((ext_vector_type(8)));

namespace {
constexpr int   kH     = 4096;            // hidden dim
constexpr int   kRank  = 16;              // LoRA rank
constexpr int   kTokPerAdapter = 4096;    // (B/E) * S = 2 * 2048 tokens per adapter
constexpr int   kTM    = 64;              // tokens per block (4 waves x 16-token tile)
constexpr int   kKC    = 64;              // H-chunk staged per iteration
constexpr int   kLda   = 68;              // LDS row stride (floats): 272B = 16B-aligned, conflict-free
constexpr int   kLdm   = 18;              // LDS stride for mid staging
constexpr float kScale = 2.0f / 0.95f;    // SCALING / (1 - P_DROP)
constexpr float kPDrop = 0.05f;
}

__global__ __launch_bounds__(128) void lora_fused_wmma(
    const float* __restrict__ result,
    const float* __restrict__ data,
    const float* __restrict__ mask,
    const float* __restrict__ lora_a,   // [E, r, H]
    const float* __restrict__ lora_b,   // [E, H, r]
    float* __restrict__ out)
{
    __shared__ float sDrop[kTM * kLda];      // dropped data tile   (64 x 64, padded)
    __shared__ float sA[kRank * kLda];       // lora_a chunk        (16 x 64, padded)
    __shared__ float sMid[kTM * kLdm];       // mid layout shuffle  (64 x 16, padded)

    const int tid  = threadIdx.x;
    const int lane = tid & 31;
    const int w    = tid >> 5;     // wave id 0..3 -> 16-token tile
    const int half = lane >> 4;    // 0: K/M-low half, 1: high half (per ISA frag layout)
    const int lr   = lane & 15;

    const int tokBase = blockIdx.x * kTM;
    const int e       = tokBase / kTokPerAdapter;    // adapter for this whole block
    const float* __restrict__ A  = lora_a + (size_t)e * kRank * kH;   // [r=16][H]
    const float* __restrict__ Bw = lora_b + (size_t)e * kH * kRank;   // [H][r=16]

    // ---------------- GEMM1: mid[16 tok x 16 r] = dropped @ A^T ----------------
    v8f mid = {};

    for (int h0 = 0; h0 < kH; h0 += kKC) {
        // Stage dropped(data,mask) tile: 64 tokens x 64 h, coalesced float4, NT (streamed once)
        #pragma unroll
        for (int t = 0; t < 8; ++t) {
            int idx = tid + t * 128;            // 0..1023
            int row = idx >> 4;                 // token row 0..63
            int c   = (idx & 15) << 2;          // h col 0..60 step 4
            size_t g = (size_t)(tokBase + row) * kH + h0 + c;
            v4f d = __builtin_nontemporal_load((const v4f*)(data + g));
            v4f m = __builtin_nontemporal_load((const v4f*)(mask + g));
            v4f r;
            r.x = (m.x >= kPDrop) ? d.x * kScale : 0.0f;
            r.y = (m.y >= kPDrop) ? d.y * kScale : 0.0f;
            r.z = (m.z >= kPDrop) ? d.z * kScale : 0.0f;
            r.w = (m.w >= kPDrop) ? d.w * kScale : 0.0f;
            *(v4f*)&sDrop[row * kLda + c] = r;
        }
        // Stage lora_a chunk: 16 r x 64 h (small, L2-resident)
        #pragma unroll
        for (int t = 0; t < 2; ++t) {
            int idx = tid + t * 128;            // 0..255
            int row = idx >> 4;                 // r row 0..15
            int c   = (idx & 15) << 2;
            v4f a = *(const v4f*)(A + (size_t)row * kH + h0 + c);
            *(v4f*)&sA[row * kLda + c] = a;
        }
        __syncthreads();

        // 16 k-steps of V_WMMA_F32_16X16X4_F32.
        // f32 A-frag 16x4 (ISA 7.12.2): lane<16 holds K={0,1}, lane>=16 holds K={2,3}
        //  -> one aligned float2 LDS load per operand per step, conflict-free (stride 68).
        const float* ap = &sDrop[(w * 16 + lr) * kLda + 2 * half];
        const float* bp = &sA[lr * kLda + 2 * half];
        #pragma unroll
        for (int kk = 0; kk < kKC / 4; ++kk) {
            v2f afrag = *(const v2f*)(ap + kk * 4);
            v2f bfrag = *(const v2f*)(bp + kk * 4);
            mid = __builtin_amdgcn_wmma_f32_16x16x4_f32(
                false, afrag, false, bfrag, (short)0, mid, false, false);
        }
        __syncthreads();
    }

    // -------- convert mid from C/D layout to A-frag layout via LDS (once) --------
    // C/D layout: lane = N + 16*(M/8), vgpr v holds M = v + 8*(lane/16), N = lane%16
    #pragma unroll
    for (int v = 0; v < 8; ++v) {
        int M = v + 8 * half;
        sMid[(w * 16 + M) * kLdm + lr] = mid[v];
    }
    __syncthreads();

    v2f am[4];
    #pragma unroll
    for (int q = 0; q < 4; ++q)
        am[q] = *(const v2f*)&sMid[(w * 16 + lr) * kLdm + q * 4 + 2 * half];

    // ---------------- GEMM2: out = result + mid @ B^T ----------------
    const int rowTok = tokBase + w * 16;
    for (int d0 = 0; d0 < kH; d0 += 16) {
        // Load result tile directly as WMMA C operand (free accumulate)
        v8f acc;
        #pragma unroll
        for (int v = 0; v < 8; ++v) {
            size_t g = (size_t)(rowTok + v + 8 * half) * kH + d0 + lr;
            acc[v] = __builtin_nontemporal_load(result + g);
        }
        // B-frag [k=r][n=d]: lora_b[e][d0+n][r]; lane n=lr, k = 2*half + {0,1} (+4q)
        #pragma unroll
        for (int q = 0; q < 4; ++q) {
            v2f bfrag = *(const v2f*)(Bw + (size_t)(d0 + lr) * kRank + q * 4 + 2 * half);
            acc = __builtin_amdgcn_wmma_f32_16x16x4_f32(
                false, am[q], false, bfrag, (short)0, acc, false, false);
        }
        #pragma unroll
        for (int v = 0; v < 8; ++v) {
            size_t g = (size_t)(rowTok + v + 8 * half) * kH + d0 + lr;
            __builtin_nontemporal_store(acc[v], out + g);
        }
    }
}

extern "C" void kernel_launch(void* const* d_in, const int* in_sizes, int n_in,
                              void* d_out, int out_size, void* d_ws, size_t ws_size,
                              hipStream_t stream) {
    const float* result    = (const float*)d_in[0];
    const float* data      = (const float*)d_in[1];
    const float* drop_mask = (const float*)d_in[2];
    const float* lora_a    = (const float*)d_in[3];
    const float* lora_b    = (const float*)d_in[4];
    float* out = (float*)d_out;

    const int numTokens = in_sizes[1] / kH;       // B*S = 32768
    const int blocks    = numTokens / kTM;        // 512
    lora_fused_wmma<<<blocks, 128, 0, stream>>>(result, data, drop_mask,
                                                lora_a, lora_b, out);
}